// NeuronAttentionBase_46402826666320
// MI455X (gfx1250) — compile-verified
//
#include <hip/hip_runtime.h>
#include <hip/hip_bf16.h>
#include <math.h>

// Problem constants (match reference)
#define BB   8
#define SCC  8192
#define HH   4096
#define NHH  32
#define NKVV 8
#define DD   128
#define GG   (NHH / NKVV)            // 4
#define QKVN ((NHH + 2 * NKVV) * DD) // 6144

// GEMM split-K factor
#define KSPLIT 4

// attention partial tiling
#define CHUNKS        8              // SC split into 8 chunks of 1024
#define WAVES_PER_BLK 8              // 256 threads
#define POS_PER_WAVE  128
#define SLOTS         (CHUNKS * WAVES_PER_BLK)  // 64 partials per (b,kv)
#define SLOT_F        (8 + GG * DD)             // m[4], l[4], acc[4][128] = 520 floats

typedef float v2f __attribute__((ext_vector_type(2)));
typedef float v8f __attribute__((ext_vector_type(8)));

// ---------------------------------------------------------------------------
// Cpart(split, 8 x N) = Apad(16 x K)[, ksplit] @ W(K x N), fp32 WMMA.
// A is zero-padded to 16 rows so fragment loads are unconditional b64 loads.
// One wave per (16-col tile, k-split). Pointer-bumped addressing.
// A-frag (16x4): lanes 0-15 hold M=lane, K={k0,k0+1}; lanes 16-31 K={k0+2,k0+3}.
// B-frag (4x16): VGPR j: lanes 0-15 row k0+j, lanes 16-31 row k0+2+j, col=lane&15.
// D (16x16): VGPR r: lanes 0-15 -> row r, lanes 16-31 -> row r+8.
// ---------------------------------------------------------------------------
__global__ void gemm8_wmma_f32(const float* __restrict__ Apad,
                               const float* __restrict__ W,
                               float* __restrict__ Cpart,
                               int K, int N) {
  const int n0    = blockIdx.x * 16;
  const int split = blockIdx.y;
  const int klen  = K / KSPLIT;
  const int kbeg  = split * klen;

  const int lane = threadIdx.x & 31;
  const int row  = lane & 15;
  const int koff = (lane >> 4) << 1;     // 0 or 2
  const int col  = n0 + (lane & 15);

  const float* ap = Apad + (size_t)row * K + kbeg + koff;      // 8B aligned
  const float* wp = W + (size_t)(kbeg + koff) * N + col;
  const size_t wstep = (size_t)4 * N;

  v8f acc = {};
#pragma unroll 8
  for (int k = 0; k < klen; k += 4) {
    const v2f a = *(const v2f*)ap;       // global_load_b64
    v2f b;
    b.x = wp[0];
    b.y = wp[(size_t)N];
    ap += 4;
    wp += wstep;
    acc = __builtin_amdgcn_wmma_f32_16x16x4_f32(
        /*neg_a=*/false, a, /*neg_b=*/false, b,
        /*c_mod=*/(short)0, acc, /*reuse_a=*/false, /*reuse_b=*/false);
  }

  if (lane < 16) {
    float* cp = Cpart + (size_t)split * BB * N + n0 + lane;
#pragma unroll
    for (int r = 0; r < BB; ++r) {
      cp[(size_t)r * N] = acc[r];
    }
  }
}

// Deterministic split-K combine: C[i] = sum over KSPLIT partials.
__global__ void splitk_reduce(const float* __restrict__ part,
                              float* __restrict__ C, int N) {
  const int idx = blockIdx.x * blockDim.x + threadIdx.x;
  const size_t tot = (size_t)BB * N;
  if (idx < (int)tot) {
    float s = 0.0f;
#pragma unroll
    for (int p = 0; p < KSPLIT; ++p) s += part[(size_t)p * tot + idx];
    C[idx] = s;
  }
}

// ---------------------------------------------------------------------------
// RoPE on q and k, scatter new k/v rows into the output caches at `pos`.
// grid: B * (NH + 2*NKV) blocks, 64 threads (each handles a rotation pair).
// ---------------------------------------------------------------------------
__global__ void rope_cache_kernel(const float* __restrict__ qkv,
                                  const int* __restrict__ pos_p,
                                  float* __restrict__ qrot,
                                  float* __restrict__ kcache,
                                  float* __restrict__ vcache) {
  const int pos = *pos_p;
  const int idx = blockIdx.x;
  const int b   = idx / (NHH + 2 * NKVV);
  const int h   = idx % (NHH + 2 * NKVV);
  const int t   = threadIdx.x;          // 0..63

  const float inv_freq = __powf(10000.0f, -(float)t / 64.0f);
  const float ang = (float)pos * inv_freq;
  const float c = __cosf(ang);
  const float s = __sinf(ang);

  const float* base;
  if (h < NHH) {
    base = qkv + (size_t)b * QKVN + (size_t)h * DD;
    const float x = base[t], y = base[t + 64];
    float* dst = qrot + ((size_t)b * NHH + h) * DD;
    dst[t]      = x * c - y * s;
    dst[t + 64] = y * c + x * s;
  } else if (h < NHH + NKVV) {
    const int kv = h - NHH;
    base = qkv + (size_t)b * QKVN + (size_t)(NHH + kv) * DD;
    const float x = base[t], y = base[t + 64];
    float* dst = kcache + (((size_t)b * NKVV + kv) * SCC + pos) * DD;
    dst[t]      = x * c - y * s;
    dst[t + 64] = y * c + x * s;
  } else {
    const int kv = h - NHH - NKVV;
    base = qkv + (size_t)b * QKVN + (size_t)(NHH + NKVV + kv) * DD;
    float* dst = vcache + (((size_t)b * NKVV + kv) * SCC + pos) * DD;
    dst[t]      = base[t];
    dst[t + 64] = base[t + 64];
  }
}

// ---------------------------------------------------------------------------
// Flash-style attention partials. grid = (CHUNKS, NKV, B), block = 256 (8 waves).
// Each wave: 128 contiguous key positions, all 4 grouped q-heads.
// Lane owns dims d = lane + 32*j (j=0..3) -> fully coalesced 128B K/V row loads.
// Prefetch 8 rows (4 KiB) ahead to keep the HBM stream saturated.
// ---------------------------------------------------------------------------
__global__ void attn_partial_kernel(const float* __restrict__ qrot,
                                    const float* __restrict__ kcache,
                                    const float* __restrict__ vcache,
                                    const int* __restrict__ pos_p,
                                    float* __restrict__ partial) {
  const int pos   = *pos_p;
  const int chunk = blockIdx.x;
  const int kv    = blockIdx.y;
  const int b     = blockIdx.z;
  const int wave  = threadIdx.x >> 5;
  const int lane  = threadIdx.x & 31;
  const int s0    = chunk * (POS_PER_WAVE * WAVES_PER_BLK) + wave * POS_PER_WAVE;
  const float scale = 0.088388347648318447f;  // 1/sqrt(128)

  // q fragments: q[h][j] = q[head kv*G+h][lane + 32*j]
  float q[GG][4];
  const float* qb = qrot + ((size_t)b * NHH + (size_t)kv * GG) * DD;
#pragma unroll
  for (int h = 0; h < GG; ++h)
#pragma unroll
    for (int j = 0; j < 4; ++j)
      q[h][j] = qb[(size_t)h * DD + lane + 32 * j];

  float m[GG], l[GG], acc[GG][4];
#pragma unroll
  for (int h = 0; h < GG; ++h) {
    m[h] = -1e30f; l[h] = 0.0f;
#pragma unroll
    for (int j = 0; j < 4; ++j) acc[h][j] = 0.0f;
  }

  const float* kb = kcache + ((size_t)b * NKVV + kv) * SCC * DD;
  const float* vb = vcache + ((size_t)b * NKVV + kv) * SCC * DD;

  for (int i = 0; i < POS_PER_WAVE; ++i) {
    const int s = s0 + i;
    if (s > pos) break;

    // prefetch 8 rows ahead (clamped to valid range) -> global_prefetch_b8
    const int sp = (s + 8 <= pos) ? (s + 8) : pos;
    __builtin_prefetch(kb + (size_t)sp * DD + lane, 0, 1);
    __builtin_prefetch(vb + (size_t)sp * DD + lane, 0, 1);

    const float* kp = kb + (size_t)s * DD;
    const float* vp = vb + (size_t)s * DD;
    const float k0 = kp[lane],      k1 = kp[lane + 32];
    const float k2 = kp[lane + 64], k3 = kp[lane + 96];
    const float v0 = vp[lane],      v1 = vp[lane + 32];
    const float v2 = vp[lane + 64], v3 = vp[lane + 96];

#pragma unroll
    for (int h = 0; h < GG; ++h) {
      float p = q[h][0] * k0 + q[h][1] * k1 + q[h][2] * k2 + q[h][3] * k3;
      // wave32 xor-reduce; every lane ends with the full dot product
#pragma unroll
      for (int off = 16; off > 0; off >>= 1)
        p += __shfl_xor(p, off, 32);
      const float sc = p * scale;
      const float nm = fmaxf(m[h], sc);
      const float f  = __expf(m[h] - nm);
      const float e  = __expf(sc - nm);
      l[h] = l[h] * f + e;
      acc[h][0] = acc[h][0] * f + e * v0;
      acc[h][1] = acc[h][1] * f + e * v1;
      acc[h][2] = acc[h][2] * f + e * v2;
      acc[h][3] = acc[h][3] * f + e * v3;
      m[h] = nm;
    }
  }

  const int slot = chunk * WAVES_PER_BLK + wave;
  float* pp = partial + (((size_t)b * NKVV + kv) * SLOTS + slot) * SLOT_F;
  if (lane == 0) {
#pragma unroll
    for (int h = 0; h < GG; ++h) { pp[h] = m[h]; pp[GG + h] = l[h]; }
  }
#pragma unroll
  for (int h = 0; h < GG; ++h)
#pragma unroll
    for (int j = 0; j < 4; ++j)
      pp[8 + h * DD + lane + 32 * j] = acc[h][j];
}

// ---------------------------------------------------------------------------
// Combine partials -> context rows (writes rows 0..7 of the padded A buffer
// for the o-projection). grid = B*NKV blocks, 128 threads (d).
// ---------------------------------------------------------------------------
__global__ void attn_reduce_kernel(const float* __restrict__ partial,
                                   float* __restrict__ attn) {
  const int bk = blockIdx.x;           // b*NKV + kv
  const int b  = bk / NKVV;
  const int kv = bk % NKVV;
  const int d  = threadIdx.x;          // 0..127
  const float* base = partial + (size_t)bk * SLOTS * SLOT_F;

#pragma unroll
  for (int h = 0; h < GG; ++h) {
    float M = -1e30f;
    for (int s = 0; s < SLOTS; ++s)
      M = fmaxf(M, base[(size_t)s * SLOT_F + h]);
    float L = 0.0f, ctx = 0.0f;
    for (int s = 0; s < SLOTS; ++s) {
      const float* pp = base + (size_t)s * SLOT_F;
      const float w = __expf(pp[h] - M);
      L   += pp[GG + h] * w;
      ctx += pp[8 + h * DD + d] * w;
    }
    attn[(size_t)b * (NHH * DD) + ((size_t)kv * GG + h) * DD + d] = ctx / L;
  }
}

extern "C" void kernel_launch(void* const* d_in, const int* in_sizes, int n_in,
                              void* d_out, int out_size, void* d_ws, size_t ws_size,
                              hipStream_t stream) {
  const float* hidden     = (const float*)d_in[0];
  const float* past_key   = (const float*)d_in[1];
  const float* past_value = (const float*)d_in[2];
  const float* w_qkv      = (const float*)d_in[3];
  const float* w_o        = (const float*)d_in[4];
  const int*   pos        = (const int*)d_in[5];

  float* out       = (float*)d_out;                          // 8*4096
  float* key_cache = out + (size_t)BB * HH;                  // 8*8*8192*128
  float* val_cache = key_cache + (size_t)BB * NKVV * SCC * DD;

  float* ws       = (float*)d_ws;
  float* hpad     = ws;                                      // 16*4096 (A for qkv gemm)
  float* qkv      = hpad    + (size_t)16 * HH;               // 8*6144
  float* qkvpart  = qkv     + (size_t)BB * QKVN;             // 4*8*6144
  float* qrot     = qkvpart + (size_t)KSPLIT * BB * QKVN;    // 8*32*128
  float* attnpad  = qrot    + (size_t)BB * NHH * DD;         // 16*4096 (A for o gemm)
  float* opart    = attnpad + (size_t)16 * HH;               // 4*8*4096
  float* partial  = opart   + (size_t)KSPLIT * BB * HH;      // 64*64*520

  const size_t cache_bytes = (size_t)BB * NKVV * SCC * DD * sizeof(float);

  // 1. Pass-through copies of the caches (updated rows overwritten below).
  hipMemcpyAsync(key_cache, past_key,   cache_bytes, hipMemcpyDeviceToDevice, stream);
  hipMemcpyAsync(val_cache, past_value, cache_bytes, hipMemcpyDeviceToDevice, stream);

  // 2. Zero-padded A operands (rows 8..15 = 0) so WMMA loads are unconditional.
  hipMemcpyAsync(hpad, hidden, (size_t)BB * HH * sizeof(float),
                 hipMemcpyDeviceToDevice, stream);
  hipMemsetAsync(hpad + (size_t)BB * HH, 0, (size_t)BB * HH * sizeof(float), stream);
  hipMemsetAsync(attnpad, 0, (size_t)16 * HH * sizeof(float), stream);

  // 3. qkv = hidden @ w_qkv  (WMMA f32, split-K x4, deterministic combine)
  gemm8_wmma_f32<<<dim3(QKVN / 16, KSPLIT), 32, 0, stream>>>(
      hpad, w_qkv, qkvpart, HH, QKVN);
  splitk_reduce<<<(BB * QKVN + 255) / 256, 256, 0, stream>>>(qkvpart, qkv, QKVN);

  // 4. RoPE q/k + write new k/v rows into caches at pos
  rope_cache_kernel<<<BB * (NHH + 2 * NKVV), 64, 0, stream>>>(
      qkv, pos, qrot, key_cache, val_cache);

  // 5. Flash attention partials over the updated caches
  attn_partial_kernel<<<dim3(CHUNKS, NKVV, BB), WAVES_PER_BLK * 32, 0, stream>>>(
      qrot, key_cache, val_cache, pos, partial);

  // 6. Reduce partials -> context rows (into padded A buffer)
  attn_reduce_kernel<<<BB * NKVV, DD, 0, stream>>>(partial, attnpad);

  // 7. out = attn @ w_o  (WMMA f32, split-K x4, deterministic combine)
  gemm8_wmma_f32<<<dim3(HH / 16, KSPLIT), 32, 0, stream>>>(
      attnpad, w_o, opart, NHH * DD, HH);
  splitk_reduce<<<(BB * HH + 255) / 256, 256, 0, stream>>>(opart, out, HH);
}